// BitterGraphSAGE_Baseline_52475910422832
// MI455X (gfx1250) — compile-verified
//
#include <hip/hip_runtime.h>

typedef float v2f __attribute__((ext_vector_type(2)));
typedef float v8f __attribute__((ext_vector_type(8)));

#define N_NODES   100000
#define N_EDGES   800000
#define IN_FEAT   20
#define HIDDEN    128
#define N_GRAPHS  512
#define N_CLASSES 2

// ---------------------------------------------------------------- degrees
__global__ void deg_kernel(const int* __restrict__ dst, float* __restrict__ deg) {
    int e = blockIdx.x * blockDim.x + threadIdx.x;
    if (e < N_EDGES) atomicAdd(&deg[dst[e]], 1.0f);
}

// ------------------------------------------------- scatter-add (float4 chunks)
template <int F>
__global__ void scatter_kernel(const int* __restrict__ src, const int* __restrict__ dst,
                               const float* __restrict__ feat, float* __restrict__ agg) {
    constexpr int C = F / 4;                    // float4 chunks per edge
    int tid = blockIdx.x * blockDim.x + threadIdx.x;
    if (tid >= N_EDGES * C) return;
    int e = tid / C;
    int c = tid % C;
    int s = src[e], d = dst[e];
    float4 v = *(const float4*)(feat + s * F + c * 4);
    float* p = agg + d * F + c * 4;
    atomicAdd(p + 0, v.x);
    atomicAdd(p + 1, v.y);
    atomicAdd(p + 2, v.z);
    atomicAdd(p + 3, v.w);
}

// ------------------------------------------------------------- SAGE GEMM
// out[row, :] = (agg[row,:] / max(deg,1)) @ Wl + bl + self[row,:] @ Wr  (opt. relu)
// One wave per 16-row tile x full 128-wide strip (8 col tiles), fp32 WMMA 16x16x4.
// W staged in LDS pre-swizzled by k-pair: sW[pair*288 + col*2 + (k&1)], so each
// lane's B fragment is ONE aligned ds_load_b64 (no repack movs). Stride 288
// (=256+32 pad) puts consecutive pairs 32 banks apart -> the two half-waves
// (which read pairs p and p+1) land on disjoint bank halves: conflict-free.
template <int K, bool RELU>
__global__ __launch_bounds__(256) void sage_gemm(
    const float* __restrict__ agg, const float* __restrict__ deg,
    const float* __restrict__ selfF,
    const float* __restrict__ Wl, const float* __restrict__ bl,
    const float* __restrict__ Wr,
    float* __restrict__ out) {
    constexpr int KC = (K < 32) ? K : 32;       // K-chunk staged in LDS
    constexpr int NPAIR = KC / 2;
    constexpr int PSTR = 2 * HIDDEN + 32;       // 288 floats: bank-padded pair stride
    __shared__ float sWl[NPAIR * PSTR];
    __shared__ float sWr[NPAIR * PSTR];

    const int lane  = threadIdx.x & 31;
    const int wave  = threadIdx.x >> 5;
    const int mtile = blockIdx.x * (blockDim.x >> 5) + wave;
    const bool active = (mtile * 16 < N_NODES); // wave-uniform => EXEC stays all-1s
    const int  row  = mtile * 16 + (lane & 15);
    const int  kh   = (lane >> 4) * 2;          // this half-wave holds K {kh, kh+1}
    const int  col0 = lane & 15;

    float inv = 1.0f;
    const float* Arow = agg;
    const float* Xrow = selfF;
    if (active) {
        inv  = 1.0f / fmaxf(deg[row], 1.0f);
        Arow = agg   + (size_t)row * K;
        Xrow = selfF + (size_t)row * K;
    }

    v8f acc[8] = {};

    for (int k0 = 0; k0 < K; k0 += KC) {
        __syncthreads();
        // Swizzled staging: one b64 store per (pair, col) slot.
        for (int i = threadIdx.x; i < NPAIR * HIDDEN; i += 256) {
            int p = i >> 7;                     // i / HIDDEN
            int col = i & (HIDDEN - 1);
            int kg = k0 + p * 2;
            v2f wl, wr;
            wl.x = Wl[(size_t)kg * HIDDEN + col];
            wl.y = Wl[(size_t)(kg + 1) * HIDDEN + col];
            wr.x = Wr[(size_t)kg * HIDDEN + col];
            wr.y = Wr[(size_t)(kg + 1) * HIDDEN + col];
            *(v2f*)&sWl[p * PSTR + col * 2] = wl;
            *(v2f*)&sWr[p * PSTR + col * 2] = wr;
        }
        __syncthreads();
        if (active) {
            for (int kk = 0; kk < KC; kk += 4) {
                v2f a1, a2;
                a1.x = Arow[k0 + kk + kh]     * inv;
                a1.y = Arow[k0 + kk + kh + 1] * inv;
                a2.x = Xrow[k0 + kk + kh];
                a2.y = Xrow[k0 + kk + kh + 1];
                // pair index this half-wave consumes: kk/2 (lanes 0-15) or kk/2+1
                const int p = (kk >> 1) + (lane >> 4);
                const float* bL = &sWl[p * PSTR + col0 * 2];
                const float* bR = &sWr[p * PSTR + col0 * 2];
                #pragma unroll
                for (int n = 0; n < 8; ++n) {
                    const v2f b1 = *(const v2f*)(bL + n * 32);  // col tile n: +16 cols
                    const v2f b2 = *(const v2f*)(bR + n * 32);
                    acc[n] = __builtin_amdgcn_wmma_f32_16x16x4_f32(
                        false, a1, false, b1, (short)0, acc[n], false, false);
                    acc[n] = __builtin_amdgcn_wmma_f32_16x16x4_f32(
                        false, a2, false, b2, (short)0, acc[n], false, false);
                }
            }
        }
    }

    if (!active) return;
    // D layout: VGPR i -> M = i (lanes 0-15) / M = 8+i (lanes 16-31), N = lane&15
    const int rbase = mtile * 16 + (lane >> 4) * 8;
    #pragma unroll
    for (int n = 0; n < 8; ++n) {
        const float b = bl[n * 16 + col0];
        #pragma unroll
        for (int i = 0; i < 8; ++i) {
            float v = acc[n][i] + b;
            if (RELU) v = fmaxf(v, 0.0f);
            out[(size_t)(rbase + i) * HIDDEN + n * 16 + col0] = v;
        }
    }
}

// ---------------------------------------------------------------- pooling
__global__ void pool_kernel(const float* __restrict__ h, const int* __restrict__ batch,
                            float* __restrict__ pooled, float* __restrict__ gcnt) {
    int tid = blockIdx.x * blockDim.x + threadIdx.x;
    if (tid >= N_NODES * (HIDDEN / 4)) return;
    int node = tid >> 5;               // HIDDEN/4 == 32 chunks
    int c    = tid & 31;
    int g    = batch[node];
    float4 v = *(const float4*)(h + (size_t)node * HIDDEN + c * 4);
    float* p = pooled + (size_t)g * HIDDEN + c * 4;
    atomicAdd(p + 0, v.x);
    atomicAdd(p + 1, v.y);
    atomicAdd(p + 2, v.z);
    atomicAdd(p + 3, v.w);
    if (c == 0) atomicAdd(&gcnt[g], 1.0f);
}

// ------------------------------------------------------------ linear head
__global__ void final_kernel(const float* __restrict__ pooled, const float* __restrict__ gcnt,
                             const float* __restrict__ Wlin, const float* __restrict__ blin,
                             float* __restrict__ out) {
    int tid = blockIdx.x * blockDim.x + threadIdx.x;
    if (tid >= N_GRAPHS * N_CLASSES) return;
    int g = tid / N_CLASSES, c = tid % N_CLASSES;
    float invn = 1.0f / fmaxf(gcnt[g], 1.0f);
    float s = 0.0f;
    #pragma unroll 4
    for (int f = 0; f < HIDDEN; ++f)
        s += pooled[g * HIDDEN + f] * Wlin[f * N_CLASSES + c];
    out[tid] = s * invn + blin[c];     // mean folded as row-constant post-scale
}

extern "C" void kernel_launch(void* const* d_in, const int* in_sizes, int n_in,
                              void* d_out, int out_size, void* d_ws, size_t ws_size,
                              hipStream_t stream) {
    const float* x    = (const float*)d_in[0];
    const int*   ei   = (const int*)d_in[1];
    const int*   bat  = (const int*)d_in[2];
    const float* Wl1  = (const float*)d_in[3];
    const float* bl1  = (const float*)d_in[4];
    const float* Wr1  = (const float*)d_in[5];
    const float* Wl2  = (const float*)d_in[6];
    const float* bl2  = (const float*)d_in[7];
    const float* Wr2  = (const float*)d_in[8];
    const float* Wl3  = (const float*)d_in[9];
    const float* bl3  = (const float*)d_in[10];
    const float* Wr3  = (const float*)d_in[11];
    const float* Wlin = (const float*)d_in[12];
    const float* blin = (const float*)d_in[13];
    const int* src = ei;                // edge_index row 0
    const int* dst = ei + N_EDGES;      // edge_index row 1

    float* ws     = (float*)d_ws;
    float* deg    = ws;                                   // N
    float* agg    = deg + N_NODES;                        // N*128 (layer1 uses N*20)
    float* h1     = agg + (size_t)N_NODES * HIDDEN;       // N*128
    float* h2     = h1  + (size_t)N_NODES * HIDDEN;       // N*128
    float* pooled = h2  + (size_t)N_NODES * HIDDEN;       // 512*128
    float* gcnt   = pooled + N_GRAPHS * HIDDEN;           // 512 (contiguous after pooled)
    float* out    = (float*)d_out;

    const int GEMM_BLOCKS = (N_NODES / 16 + 7) / 8;       // 8 waves (m-tiles) per block

    // degrees (shared by all layers)
    hipMemsetAsync(deg, 0, N_NODES * sizeof(float), stream);
    deg_kernel<<<(N_EDGES + 255) / 256, 256, 0, stream>>>(dst, deg);

    // ---- layer 1: x[N,20] -> h1[N,128], relu
    hipMemsetAsync(agg, 0, (size_t)N_NODES * IN_FEAT * sizeof(float), stream);
    scatter_kernel<IN_FEAT><<<(N_EDGES * (IN_FEAT / 4) + 255) / 256, 256, 0, stream>>>(src, dst, x, agg);
    sage_gemm<IN_FEAT, true><<<GEMM_BLOCKS, 256, 0, stream>>>(agg, deg, x, Wl1, bl1, Wr1, h1);

    // ---- layer 2: h1 -> h2, relu
    hipMemsetAsync(agg, 0, (size_t)N_NODES * HIDDEN * sizeof(float), stream);
    scatter_kernel<HIDDEN><<<(N_EDGES * (HIDDEN / 4) + 255) / 256, 256, 0, stream>>>(src, dst, h1, agg);
    sage_gemm<HIDDEN, true><<<GEMM_BLOCKS, 256, 0, stream>>>(agg, deg, h1, Wl2, bl2, Wr2, h2);

    // ---- layer 3: h2 -> h3 (reuse h1), no relu
    hipMemsetAsync(agg, 0, (size_t)N_NODES * HIDDEN * sizeof(float), stream);
    scatter_kernel<HIDDEN><<<(N_EDGES * (HIDDEN / 4) + 255) / 256, 256, 0, stream>>>(src, dst, h2, agg);
    sage_gemm<HIDDEN, false><<<GEMM_BLOCKS, 256, 0, stream>>>(agg, deg, h2, Wl3, bl3, Wr3, h1);

    // ---- global mean pool + linear head
    hipMemsetAsync(pooled, 0, (N_GRAPHS * HIDDEN + N_GRAPHS) * sizeof(float), stream);
    pool_kernel<<<(N_NODES * (HIDDEN / 4) + 255) / 256, 256, 0, stream>>>(h1, bat, pooled, gcnt);
    final_kernel<<<(N_GRAPHS * N_CLASSES + 255) / 256, 256, 0, stream>>>(pooled, gcnt, Wlin, blin, out);
}